// MemeticCommAdapter_23725399343645
// MI455X (gfx1250) — compile-verified
//
#include <hip/hip_runtime.h>

typedef float v2f __attribute__((ext_vector_type(2)));
typedef float v8f __attribute__((ext_vector_type(8)));

#define NB   4096
#define NAG  64
#define HD   256
#define UD   128
#define ZD   16
#define AD   64
#define SCALE  0.25f
#define LN_EPS 1e-5f

// LDS strides (floats) -- padded so (stride*m + k) mod 64 is conflict-free
#define SH 260   // h rows [64][256]
#define SU 132   // u / c rows [64][128]
#define SZ 20    // z rows [64][16]
#define SA 68    // q/k/v/scores/c_latent rows [64][64]
#define STR 4    // t (rank) rows [64][4]
#define SG 20    // gate/cand rows [64][16]

// LDS float offsets
#define OFF_H  0
#define OFF_UC (OFF_H + 64*SH)          // u, later c
#define OFF_Z  (OFF_UC + 64*SU)
#define OFF_K  (OFF_Z + 64*SZ)          // k, later c_latent
#define OFF_V  (OFF_K + 64*SA)          // v, later gate+cand
#define OFF_Q  (OFF_V + 64*SA)          // q, later scores/attn
#define OFF_T  (OFF_Q + 64*SA)          // 3 x [64][4]
#define LDS_FLOATS (OFF_T + 3*256)      // 40192 floats = 160768 bytes

__device__ __forceinline__ v8f wmma_f32(v2f a, v2f b, v8f c) {
  // (neg_a, A, neg_b, B, c_mod, C, reuse_a, reuse_b)
  return __builtin_amdgcn_wmma_f32_16x16x4_f32(false, a, false, b, (short)0, c,
                                               false, false);
}

// 16B async copy global -> LDS (CDNA5 GLOBAL_LOAD_ASYNC_TO_LDS_B128, ASYNCcnt)
__device__ __forceinline__ void async_cp16(unsigned lds_byte, const void* gptr) {
  asm volatile("global_load_async_to_lds_b128 %0, %1, off"
               :: "v"(lds_byte), "v"(gptr)
               : "memory");
}
__device__ __forceinline__ void wait_async0() {
  asm volatile("s_wait_asynccnt 0" ::: "memory");
}

__global__ __launch_bounds__(256)
void memetic_fused_kernel(
    const float* __restrict__ h,   const float* __restrict__ u,
    const float* __restrict__ z,
    const float* __restrict__ qd_down, const float* __restrict__ qd_up,
    const float* __restrict__ kd_down, const float* __restrict__ kd_up,
    const float* __restrict__ vd_down, const float* __restrict__ vd_up,
    const float* __restrict__ od_down, const float* __restrict__ od_up,
    const float* __restrict__ ug_w,    const float* __restrict__ ug_b,
    const float* __restrict__ cand_w,  const float* __restrict__ cand_b,
    const float* __restrict__ eta_logit,
    float* __restrict__ out_c,    float* __restrict__ out_z,
    float* __restrict__ out_attn, float* __restrict__ out_q,
    float* __restrict__ out_k,    float* __restrict__ out_v)
{
  extern __shared__ float sm[];
  const int tid  = threadIdx.x;
  const int b    = blockIdx.x;
  const int lane = tid & 31;
  // wave index as an SGPR: keeps tile loops scalar (no EXEC-masked loops
  // around WMMA, which requires EXEC == all ones)
  const int wave = __builtin_amdgcn_readfirstlane(tid >> 5);
  const int l15  = lane & 15;
  const int half = lane >> 4;
  const int kb   = half * 2;          // this lane-half's K sub-slot {0,2}

  float* h_s  = sm + OFF_H;
  float* u_s  = sm + OFF_UC;
  float* c_s  = sm + OFF_UC;          // aliases u (u dead after t_k/t_v)
  float* z_s  = sm + OFF_Z;
  float* k_s  = sm + OFF_K;
  float* cl_s = sm + OFF_K;           // c_latent aliases k (k dead after scores)
  float* v_s  = sm + OFF_V;
  float* g_s  = sm + OFF_V;           // gate aliases v (v dead after attn@v)
  float* p_s  = sm + OFF_V + 64*SG;   // cand
  float* q_s  = sm + OFF_Q;
  float* sc_s = sm + OFF_Q;           // scores/attn alias q (q dead after scores)
  float* t_s  = sm + OFF_T;           // 3 rank buffers of 256 floats

  // ------------- Phase 1: async-stage h, u, z into padded LDS -------------
  {
    const unsigned lds0 = __builtin_amdgcn_groupstaticsize(); // dyn-LDS base
    const float4* hg = (const float4*)(h + (size_t)b * NAG * HD);
    for (int i = tid; i < NAG * HD / 4; i += 256) {
      int r = i >> 6, c = (i & 63) << 2;
      async_cp16(lds0 + (unsigned)(OFF_H + r * SH + c) * 4u, (const void*)(hg + i));
    }
    const float4* ug = (const float4*)(u + (size_t)b * NAG * UD);
    for (int i = tid; i < NAG * UD / 4; i += 256) {
      int r = i >> 5, c = (i & 31) << 2;
      async_cp16(lds0 + (unsigned)(OFF_UC + r * SU + c) * 4u, (const void*)(ug + i));
    }
    const float4* zg = (const float4*)(z + (size_t)b * NAG * ZD);
    for (int i = tid; i < NAG * ZD / 4; i += 256) {
      int r = i >> 2, c = (i & 3) << 2;
      async_cp16(lds0 + (unsigned)(OFF_Z + r * SZ + c) * 4u, (const void*)(zg + i));
    }
    wait_async0();
  }
  __syncthreads();

  // rank-4 up-projection [64,4]@[4,64]*scale via 16 WMMA tiles, 2 per wave
  auto up64 = [&](const float* tb, const float* __restrict__ wup,
                  float* lds_out, float* __restrict__ gout, float scale) {
#pragma unroll
    for (int i = 0; i < 2; ++i) {
      int t  = wave + 8 * i;            // scalar tile id
      int mt = t >> 2, nt = t & 3;
      int n  = nt * 16 + l15;
      v2f a;  a.x  = tb[(mt * 16 + l15) * STR + kb];
              a.y  = tb[(mt * 16 + l15) * STR + kb + 1];
      v2f bb; bb.x = wup[kb * AD + n];
              bb.y = wup[(kb + 1) * AD + n];
      v8f c = {};
      c = wmma_f32(a, bb, c);
#pragma unroll
      for (int r = 0; r < 8; ++r) {
        int m = mt * 16 + r + half * 8;
        float val = c[r] * scale;
        lds_out[m * SA + n] = val;
        gout[((size_t)b * NAG + m) * AD + n] = val;
      }
    }
  };

  // ---------------- Phase 2: q = (concat(h,z) @ qd_down) @ qd_up ----------
  {
    int row = tid >> 2, rr = tid & 3;
    float acc = 0.f;
    for (int i = 0; i < HD; ++i) acc += h_s[row * SH + i] * qd_down[i * 4 + rr];
    for (int i = 0; i < ZD; ++i) acc += z_s[row * SZ + i] * qd_down[(HD + i) * 4 + rr];
    t_s[0 * 256 + row * STR + rr] = acc;
  }
  __syncthreads();
  up64(&t_s[0], qd_up, q_s, out_q, SCALE);
  __syncthreads();

  // ---------------- Phase 3: k, v --------------------------------------
  {
    int row = tid >> 2, rr = tid & 3;
    float acc = 0.f;
    for (int i = 0; i < UD; ++i) acc += u_s[row * SU + i] * kd_down[i * 4 + rr];
    t_s[1 * 256 + row * STR + rr] = acc;
  }
  __syncthreads();
  up64(&t_s[256], kd_up, k_s, out_k, SCALE);
  __syncthreads();
  {
    int row = tid >> 2, rr = tid & 3;
    float acc = 0.f;
    for (int i = 0; i < UD; ++i) acc += u_s[row * SU + i] * vd_down[i * 4 + rr];
    t_s[2 * 256 + row * STR + rr] = acc;
  }
  __syncthreads();
  up64(&t_s[512], vd_up, v_s, out_v, SCALE);
  __syncthreads();

  // ---------------- Phase 4: scores = q @ k^T / 8 -----------------------
  {
    // wave handles tiles (mt0,nt) and (mt0+2,nt); B operand shared
    int mt0 = wave >> 2, nt = wave & 3, mt1 = mt0 + 2;
    v8f acc0 = {}, acc1 = {};
#pragma unroll 4
    for (int s = 0; s < 16; ++s) {
      int kk = s * 4;
      v2f a0, a1, bb;
      a0.x = q_s[(mt0 * 16 + l15) * SA + kk + kb];
      a0.y = q_s[(mt0 * 16 + l15) * SA + kk + kb + 1];
      a1.x = q_s[(mt1 * 16 + l15) * SA + kk + kb];
      a1.y = q_s[(mt1 * 16 + l15) * SA + kk + kb + 1];
      bb.x = k_s[(nt * 16 + l15) * SA + kk + kb];     // B[d][n] = k[n][d]
      bb.y = k_s[(nt * 16 + l15) * SA + kk + kb + 1];
      acc0 = wmma_f32(a0, bb, acc0);
      acc1 = wmma_f32(a1, bb, acc1);
    }
    __syncthreads();                      // all q_s reads done before aliasing
    const float iss = 0.125f;             // 1/sqrt(64)
#pragma unroll
    for (int r = 0; r < 8; ++r) {
      sc_s[(mt0 * 16 + r + half * 8) * SA + nt * 16 + l15] = acc0[r] * iss;
      sc_s[(mt1 * 16 + r + half * 8) * SA + nt * 16 + l15] = acc1[r] * iss;
    }
  }
  __syncthreads();

  // ---------------- Phase 5: masked softmax (4 lanes per row) ------------
  {
    int row = tid >> 2, js = (tid & 3) * 16;
    float vals[16];
    float mx = -3.0e38f;
#pragma unroll
    for (int j = 0; j < 16; ++j) {
      int col = js + j;
      float s = (col == row) ? -1e30f : sc_s[row * SA + col];
      vals[j] = s;
      mx = fmaxf(mx, s);
    }
    mx = fmaxf(mx, __shfl_xor(mx, 1, 32));
    mx = fmaxf(mx, __shfl_xor(mx, 2, 32));
    float sum = 0.f;
#pragma unroll
    for (int j = 0; j < 16; ++j) {
      float e = __expf(vals[j] - mx);     // diagonal underflows to exact 0
      vals[j] = e;
      sum += e;
    }
    sum += __shfl_xor(sum, 1, 32);
    sum += __shfl_xor(sum, 2, 32);
    float inv = 1.0f / sum;
#pragma unroll
    for (int j = 0; j < 16; ++j) {
      float a = vals[j] * inv;
      sc_s[row * SA + js + j] = a;
      out_attn[((size_t)b * NAG + row) * NAG + js + j] = a;
    }
  }
  __syncthreads();

  // ---------------- Phase 6: c_latent = attn @ v ------------------------
  {
    int mt0 = wave >> 2, nt = wave & 3, mt1 = mt0 + 2;
    v8f acc0 = {}, acc1 = {};
#pragma unroll 4
    for (int s = 0; s < 16; ++s) {
      int kk = s * 4;
      v2f a0, a1, bb;
      a0.x = sc_s[(mt0 * 16 + l15) * SA + kk + kb];
      a0.y = sc_s[(mt0 * 16 + l15) * SA + kk + kb + 1];
      a1.x = sc_s[(mt1 * 16 + l15) * SA + kk + kb];
      a1.y = sc_s[(mt1 * 16 + l15) * SA + kk + kb + 1];
      bb.x = v_s[(kk + kb) * SA + nt * 16 + l15];     // B[d][n] = v[d][n]
      bb.y = v_s[(kk + kb + 1) * SA + nt * 16 + l15];
      acc0 = wmma_f32(a0, bb, acc0);
      acc1 = wmma_f32(a1, bb, acc1);
    }
#pragma unroll
    for (int r = 0; r < 8; ++r) {   // cl_s aliases k_s (dead); distinct region
      cl_s[(mt0 * 16 + r + half * 8) * SA + nt * 16 + l15] = acc0[r];
      cl_s[(mt1 * 16 + r + half * 8) * SA + nt * 16 + l15] = acc1[r];
    }
  }
  __syncthreads();

  // ---------------- Phase 7: c = (c_latent @ od_down) @ od_up * SCALE ----
  {
    int row = tid >> 2, rr = tid & 3;
    float acc = 0.f;
    for (int i = 0; i < AD; ++i) acc += cl_s[row * SA + i] * od_down[i * 4 + rr];
    t_s[0 * 256 + row * STR + rr] = acc;
  }
  __syncthreads();
#pragma unroll
  for (int i = 0; i < 4; ++i) {            // [64,4]@[4,128]: 4x8 tiles
    int t  = wave + 8 * i;                 // scalar tile id
    int mt = t >> 3, nt = t & 7;
    int n  = nt * 16 + l15;
    v2f a;  a.x  = t_s[(mt * 16 + l15) * STR + kb];
            a.y  = t_s[(mt * 16 + l15) * STR + kb + 1];
    v2f bb; bb.x = od_up[kb * UD + n];
            bb.y = od_up[(kb + 1) * UD + n];
    v8f c = {};
    c = wmma_f32(a, bb, c);
#pragma unroll
    for (int r = 0; r < 8; ++r) {
      int m = mt * 16 + r + half * 8;
      float val = c[r] * SCALE;
      c_s[m * SU + n] = val;               // aliases u_s (dead)
      out_c[((size_t)b * NAG + m) * UD + n] = val;
    }
  }
  __syncthreads();

  // ------- Phase 8: gate/cand = act(concat(z,h,c) @ W + bias), K=400 -----
  {
    int mt  = wave & 3;                    // row tile (scalar)
    int mat = wave >> 2;                   // 0 = gate, 1 = cand (scalar)
    const float* W    = mat ? cand_w : ug_w;
    const float* bias = mat ? cand_b : ug_b;
    float* outg       = mat ? p_s : g_s;   // aliases v_s (dead)
    v8f acc = {};
#pragma unroll
    for (int s = 0; s < 4; ++s) {          // z segment: W rows 0..15
      int kk = s * 4;
      v2f a;  a.x  = z_s[(mt * 16 + l15) * SZ + kk + kb];
              a.y  = z_s[(mt * 16 + l15) * SZ + kk + kb + 1];
      v2f bb; bb.x = W[(kk + kb) * ZD + l15];
              bb.y = W[(kk + kb + 1) * ZD + l15];
      acc = wmma_f32(a, bb, acc);
    }
#pragma unroll 4
    for (int s = 0; s < 64; ++s) {         // h segment: W rows 16..271
      int kk = s * 4;
      v2f a;  a.x  = h_s[(mt * 16 + l15) * SH + kk + kb];
              a.y  = h_s[(mt * 16 + l15) * SH + kk + kb + 1];
      v2f bb; bb.x = W[(16 + kk + kb) * ZD + l15];
              bb.y = W[(16 + kk + kb + 1) * ZD + l15];
      acc = wmma_f32(a, bb, acc);
    }
#pragma unroll 4
    for (int s = 0; s < 32; ++s) {         // c segment: W rows 272..399
      int kk = s * 4;
      v2f a;  a.x  = c_s[(mt * 16 + l15) * SU + kk + kb];
              a.y  = c_s[(mt * 16 + l15) * SU + kk + kb + 1];
      v2f bb; bb.x = W[(272 + kk + kb) * ZD + l15];
              bb.y = W[(272 + kk + kb + 1) * ZD + l15];
      acc = wmma_f32(a, bb, acc);
    }
    float bv = bias[l15];
#pragma unroll
    for (int r = 0; r < 8; ++r) {
      int m = mt * 16 + r + half * 8;
      float x = acc[r] + bv;
      float act = mat ? tanhf(x) : 1.f / (1.f + __expf(-x));
      outg[m * SG + l15] = act;
    }
  }
  __syncthreads();

  // ---------------- Phase 9: z_next = LN((1-eg)*z + eg*proposal) ---------
  if (tid < NAG) {
    float eta = 1.f / (1.f + __expf(-eta_logit[0]));
    float vals[ZD];
    float mu = 0.f;
#pragma unroll
    for (int j = 0; j < ZD; ++j) {
      float g  = g_s[tid * SG + j];
      float p  = p_s[tid * SG + j];
      float eg = eta * g;
      float v  = (1.f - eg) * z_s[tid * SZ + j] + eg * p;
      vals[j] = v;
      mu += v;
    }
    mu *= (1.f / ZD);
    float var = 0.f;
#pragma unroll
    for (int j = 0; j < ZD; ++j) { float d = vals[j] - mu; var += d * d; }
    var *= (1.f / ZD);
    float inv = rsqrtf(var + LN_EPS);
#pragma unroll
    for (int j = 0; j < ZD; ++j)
      out_z[((size_t)b * NAG + tid) * ZD + j] = (vals[j] - mu) * inv;
  }
}

extern "C" void kernel_launch(void* const* d_in, const int* in_sizes, int n_in,
                              void* d_out, int out_size, void* d_ws, size_t ws_size,
                              hipStream_t stream) {
  (void)in_sizes; (void)n_in; (void)out_size; (void)d_ws; (void)ws_size;
  const float* h        = (const float*)d_in[0];
  const float* u        = (const float*)d_in[1];
  const float* z        = (const float*)d_in[2];
  const float* qd_down  = (const float*)d_in[3];
  const float* qd_up    = (const float*)d_in[4];
  const float* kd_down  = (const float*)d_in[5];
  const float* kd_up    = (const float*)d_in[6];
  const float* vd_down  = (const float*)d_in[7];
  const float* vd_up    = (const float*)d_in[8];
  const float* od_down  = (const float*)d_in[9];
  const float* od_up    = (const float*)d_in[10];
  const float* ug_w     = (const float*)d_in[11];
  const float* ug_b     = (const float*)d_in[12];
  const float* cand_w   = (const float*)d_in[13];
  const float* cand_b   = (const float*)d_in[14];
  const float* eta      = (const float*)d_in[15];

  float* o = (float*)d_out;
  const size_t BN = (size_t)NB * NAG;
  float* out_c    = o;
  float* out_z    = out_c + BN * UD;
  float* out_attn = out_z + BN * ZD;
  float* out_q    = out_attn + BN * NAG;
  float* out_k    = out_q + BN * AD;
  float* out_v    = out_k + BN * AD;

  const size_t shmem = (size_t)LDS_FLOATS * sizeof(float);  // 160768 B
  hipFuncSetAttribute(reinterpret_cast<const void*>(memetic_fused_kernel),
                      hipFuncAttributeMaxDynamicSharedMemorySize, (int)shmem);
  memetic_fused_kernel<<<NB, 256, shmem, stream>>>(
      h, u, z, qd_down, qd_up, kd_down, kd_up, vd_down, vd_up,
      od_down, od_up, ug_w, ug_b, cand_w, cand_b, eta,
      out_c, out_z, out_attn, out_q, out_k, out_v);
}